// SimilarityLayer_57337813401715
// MI455X (gfx1250) — compile-verified
//
#include <hip/hip_runtime.h>
#include <hip/hip_bf16.h>
#include <math.h>

typedef __attribute__((ext_vector_type(2))) float v2f;
typedef __attribute__((ext_vector_type(8))) float v8f;

#define NW 10        // proto rows (w)
#define NQ 300       // query rows (q)
#define C  640       // contraction dim
#define NI 196       // i / j extent
#define NT 13        // 16-wide tiles covering 196 (padded to 208)
#define KC 64        // K chunk staged in LDS (16 wmma k-steps)
#define THREADS 416  // 13 wave32s, one column tile per wave

// Swizzled LDS image: pos(tile, kc, half, n, second)
//   = ((tile*16 + kc)*32 + half*16 + n)*2 + second
// so a wave's fragment for (tile, kc) is one conflict-free ds_load_b64 at
// float2 index (tile*16 + kc)*32 + lane.
#define SWSZ (NT * 16 * 32 * 2)   // 13312 floats = 52 KB per matrix

// ---------------------------------------------------------------------------
// Pre-pass: column L2 norms. p_norm[w,i] = sqrt(sum_c p[w,c,i]^2), likewise q.
// ---------------------------------------------------------------------------
__global__ void sim_norm_kernel(const float* __restrict__ p,
                                const float* __restrict__ q,
                                float* __restrict__ pn,   // 10*196
                                float* __restrict__ qn) { // 300*196
    int idx = blockIdx.x * blockDim.x + threadIdx.x;
    const int NP = NW * NI;
    const int NQN = NQ * NI;
    if (idx < NP) {
        int w = idx / NI, i = idx % NI;
        const float* base = p + (size_t)w * C * NI + i;
        float s = 0.f;
        for (int c = 0; c < C; ++c) { float v = base[(size_t)c * NI]; s += v * v; }
        pn[idx] = sqrtf(s);
    } else if (idx < NP + NQN) {
        int k = idx - NP;
        int qi = k / NI, j = k % NI;
        const float* base = q + (size_t)qi * C * NI + j;
        float s = 0.f;
        for (int c = 0; c < C; ++c) { float v = base[(size_t)c * NI]; s += v * v; }
        qn[k] = sqrtf(s);
    }
}

// ---------------------------------------------------------------------------
// Stage one 64 x 196 chunk (rows kb..kb+63) of a c-major slab into the
// swizzled LDS image, using float4 global loads. 4-element groups never
// cross a 16-column tile boundary (4 | 16).
// ---------------------------------------------------------------------------
__device__ __forceinline__
void stage_chunk(const float* __restrict__ src, int kb, float* __restrict__ dst,
                 int tid) {
    for (int idx = tid; idx < KC * (NI / 4); idx += THREADS) {
        int c  = idx / (NI / 4);
        int g  = idx - c * (NI / 4);
        int x0 = g * 4;
        const float4 v = *(const float4*)(src + (size_t)(kb + c) * NI + x0);
        int kc = c >> 2, sub = c & 3;
        int hf = sub >> 1, second = sub & 1;
        int tile = x0 >> 4, n0 = x0 & 15;
        float* d = dst + ((tile * 16 + kc) * 32 + hf * 16 + n0) * 2 + second;
        d[0] = v.x; d[2] = v.y; d[4] = v.z; d[6] = v.w;
    }
}

// ---------------------------------------------------------------------------
// Main kernel: one block per (q,w) pair computes the full 196x196 tile.
// 13 waves; wave jt owns column tile jt and register-blocks all 13 row
// tiles (13 x v8f accumulators). Per k-step: 1 shared B fragment ds_load_b64
// + 13 A fragment ds_load_b64 feeding 13 v_wmma_f32_16x16x4_f32.
// ---------------------------------------------------------------------------
__global__ __launch_bounds__(THREADS)
void sim_wmma_kernel(const float* __restrict__ p,
                     const float* __restrict__ q,
                     const float* __restrict__ pn,
                     const float* __restrict__ qn,
                     float* __restrict__ out) {
    const int pair = blockIdx.x;           // q*10 + w
    const int qi   = pair / NW;
    const int w    = pair % NW;
    const int tid  = threadIdx.x;
    const int lane = tid & 31;
    const int jt   = tid >> 5;             // column tile 0..12
    const int hf   = lane >> 4;            // C/D: lanes 16-31 hold M = r+8
    const int n    = lane & 15;

    __shared__ float As[SWSZ];             // proto slab, swizzled, 52 KB
    __shared__ float Bs[SWSZ];             // query slab, swizzled, 52 KB
    const v2f* As2 = (const v2f*)As;
    const v2f* Bs2 = (const v2f*)Bs;

    // Zero padding columns (tile 12, n >= 4) once; never rewritten after.
    for (int idx = tid; idx < 16 * 2 * 12 * 2; idx += THREADS) {
        int kc = idx / 48;
        int r  = idx % 48;
        int h2 = r / 24; r %= 24;
        int nn = 4 + r / 2;
        int sec = r & 1;
        int off = ((12 * 16 + kc) * 32 + h2 * 16 + nn) * 2 + sec;
        As[off] = 0.f;
        Bs[off] = 0.f;
    }

    const float* pbase = p + (size_t)w  * C * NI;
    const float* qbase = q + (size_t)qi * C * NI;

    v8f acc[NT];
#pragma unroll
    for (int rt = 0; rt < NT; ++rt)
        acc[rt] = (v8f){0.f, 0.f, 0.f, 0.f, 0.f, 0.f, 0.f, 0.f};

    for (int kb = 0; kb < C; kb += KC) {
        __syncthreads();                    // protect prior reads / pad init
        stage_chunk(pbase, kb, As, tid);
        stage_chunk(qbase, kb, Bs, tid);
        __syncthreads();

        for (int kc = 0; kc < 16; ++kc) {
            const v2f b = Bs2[(jt * 16 + kc) * 32 + lane];
#pragma unroll
            for (int rt = 0; rt < NT; ++rt) {
                const v2f a = As2[(rt * 16 + kc) * 32 + lane];
                acc[rt] = __builtin_amdgcn_wmma_f32_16x16x4_f32(
                    false, a, false, b, (short)0, acc[rt], false, false);
            }
        }
    }

    // Epilogue: sim = dots / max(p_norm * q_norm, 1e-8)
    const int j = jt * 16 + n;
    const float qnv = (j < NI) ? qn[qi * NI + j] : 1.f;
    const float* pnw = pn + w * NI;
    float* obase = out + (size_t)pair * NI * NI;
    if (j < NI) {
#pragma unroll
        for (int rt = 0; rt < NT; ++rt) {
#pragma unroll
            for (int r = 0; r < 8; ++r) {
                int i = rt * 16 + r + 8 * hf;
                if (i < NI) {
                    float denom = fmaxf(pnw[i] * qnv, 1e-8f);
                    obase[(size_t)i * NI + j] = acc[rt][r] / denom;
                }
            }
        }
    }
}

// ---------------------------------------------------------------------------
extern "C" void kernel_launch(void* const* d_in, const int* in_sizes, int n_in,
                              void* d_out, int out_size, void* d_ws, size_t ws_size,
                              hipStream_t stream) {
    const float* proto = (const float*)d_in[0];   // (1,10,640,1,196)
    const float* query = (const float*)d_in[1];   // (1,300,640,1,196)
    float* out = (float*)d_out;                   // (300,10,196,196)

    float* pn = (float*)d_ws;                     // 10*196 floats
    float* qn = pn + NW * NI;                     // 300*196 floats

    const int nNorm = NW * NI + NQ * NI;          // 60760
    sim_norm_kernel<<<(nNorm + 255) / 256, 256, 0, stream>>>(proto, query, pn, qn);

    // One block per (q,w) pair, 13 wave32s each.
    sim_wmma_kernel<<<NQ * NW, THREADS, 0, stream>>>(proto, query, pn, qn, out);
}